// YoloLoss_3865470567009
// MI455X (gfx1250) — compile-verified
//
#include <hip/hip_runtime.h>

// ---- problem constants (match reference) ----
#define S_FM   26
#define A_N    5
#define C_N    80
#define B_N    64
#define T_N    16
#define N_CELL (B_N * S_FM * S_FM * A_N)   // 216320
#define IMG_N  (S_FM * S_FM * A_N)          // 3380
#define GRID_N (S_FM * S_FM)                // 676
#define DS_R   32.0f
#define IOU_THR 0.5f
#define SCALE_OBJ 5.0f

typedef __attribute__((ext_vector_type(2))) float v2f;
typedef __attribute__((ext_vector_type(8))) float v8f;

// ------------------------------------------------------------------
// Kernel 0: zero the scale-weight array, the compaction counter, and d_out.
// ------------------------------------------------------------------
__global__ void k_init(float* __restrict__ sw, int* __restrict__ counter,
                       float* __restrict__ out) {
  int n = blockIdx.x * blockDim.x + threadIdx.x;
  if (n < N_CELL) sw[n] = 0.0f;
  if (n == 0) {
    *counter = 0;
    for (int i = 0; i < 6; ++i) out[i] = 0.0f;
  }
}

// ------------------------------------------------------------------
// Kernel 1: target assignment. One thread per image, sequential over the
// T ground-truth boxes so scatter collisions resolve in loop order
// (last write wins), matching the reference's per-box loop semantics.
// sw encodes obj_mask: sw>0 <=> m==1, sw==-1 <=> m==-1, sw==0 <=> m==0.
// ------------------------------------------------------------------
__global__ void k_assign(const float* __restrict__ pred_wh,
                         const float* __restrict__ tobj,
                         const float* __restrict__ anchors,
                         float* __restrict__ sw, float* __restrict__ piou,
                         int* __restrict__ wrt) {
  int b = threadIdx.x;
  if (b >= B_N) return;
  float aw[A_N], ah[A_N];
  for (int a = 0; a < A_N; ++a) { aw[a] = anchors[2 * a]; ah[a] = anchors[2 * a + 1]; }

  for (int t = 0; t < T_N; ++t) {
    const float* g4 = tobj + (b * T_N + t) * 4;
    float gx = g4[0] / DS_R, gy = g4[1] / DS_R;
    float gw = g4[2] / DS_R, gh = g4[3] / DS_R;
    int i = (int)floorf(gx); i = min(max(i, 0), S_FM - 1);
    int j = (int)floorf(gy); j = min(max(j, 0), S_FM - 1);
    float ga = gw * gh;

    float iou_a[A_N]; int best = 0; float bv = -1.0f;
    for (int a = 0; a < A_N; ++a) {
      float inter = fminf(gw, aw[a]) * fminf(gh, ah[a]);
      iou_a[a] = inter / (ga + aw[a] * ah[a] - inter);
      if (iou_a[a] > bv) { bv = iou_a[a]; best = a; }   // argmax: first max wins
    }

    int cellBase = ((b * S_FM + i) * S_FM + j) * A_N;
    float pw = pred_wh[(cellBase + best) * 2 + 0];
    float ph = pred_wh[(cellBase + best) * 2 + 1];
    float inter = fminf(pw, gw) * fminf(ph, gh);
    float ipg = inter / (pw * ph + ga - inter);
    float sb = 2.0f - (gw / (float)S_FM) * (gh / (float)S_FM);

    for (int a = 0; a < A_N; ++a) {
      int cell = cellBase + a;
      if (a == best)              { sw[cell] = sb;   piou[cell] = ipg; }
      else if (iou_a[a] > IOU_THR){ sw[cell] = -1.0f; /* piou skipped */ }
      else                        { sw[cell] = 0.0f; piou[cell] = ipg; }
    }
    wrt[cellBase + best] = b * T_N + t;   // last best-writer (gives tb & ts row)
  }
}

// ------------------------------------------------------------------
// Kernel 2: compact active (sw>0) cells into a list (<= B*T entries).
// ------------------------------------------------------------------
__global__ void k_compact(const float* __restrict__ sw, int* __restrict__ list,
                          int* __restrict__ counter) {
  int n = blockIdx.x * blockDim.x + threadIdx.x;
  if (n >= N_CELL) return;
  if (sw[n] > 0.0f) {
    int p = atomicAdd(counter, 1);
    list[p] = n;
  }
}

// ------------------------------------------------------------------
// Kernel 3: score loss via WMMA. Each wave takes 16 active cells; the
// per-cell 80-class residual row X[m,:] = cls[m,:] - onehot_label is fed
// as both A (16x4) and B (4x16 = A^T, identical lane layout) to
// V_WMMA_F32_16X16X4_F32, accumulating X·X^T; diag(D)[m] = sum_c X[m,c]^2.
// Branch-free hot loop: out-of-range rows are clamped to a valid address
// and zeroed with a 0/1 mask, so EXEC stays all-ones across every WMMA
// and the 20 load->wmma steps can pipeline without per-step exec saves.
// ------------------------------------------------------------------
__global__ void k_score(const float* __restrict__ cls,
                        const float* __restrict__ tlab,
                        const int* __restrict__ list,
                        const int* __restrict__ counter,
                        const int* __restrict__ wrt,
                        float* __restrict__ out) {
  int lane = threadIdx.x & 31;
  int wave = (blockIdx.x * blockDim.x + threadIdx.x) >> 5;
  int count = *counter;
  int base = wave * 16;
  if (base >= count) return;          // wave-uniform exit: EXEC stays all-ones

  int r    = lane & 15;               // matrix row m handled by this lane
  int koff = (lane >> 4) * 2;         // K sub-column pair (ISA 16x4 A layout)
  int idx  = base + r;
  float vmask = (idx < count) ? 1.0f : 0.0f;
  int idx_c = min(idx, count - 1);    // count >= 1 here: always a valid address
  int cell = list[idx_c];
  int wr   = wrt[cell];
  const float* crow = cls  + (long)cell * C_N + koff;
  const float* trow = tlab + (long)wr   * C_N + koff;

  v8f acc = {0.f, 0.f, 0.f, 0.f, 0.f, 0.f, 0.f, 0.f};
#pragma unroll
  for (int k0 = 0; k0 < C_N; k0 += 4) {
    v2f a;
    a.x = (crow[k0]     - trow[k0])     * vmask;
    a.y = (crow[k0 + 1] - trow[k0 + 1]) * vmask;
    // same registers serve as A and B = A^T (mirrored 16x4 / 4x16 layouts)
    acc = __builtin_amdgcn_wmma_f32_16x16x4_f32(
        /*neg_a=*/false, a, /*neg_b=*/false, a,
        /*c_mod=*/(short)0, acc, /*reuse_a=*/false, /*reuse_b=*/false);
  }

  // Extract diag(D): D VGPR q, lanes 0-15 -> M=q, N=lane; lanes 16-31 -> M=8+q.
  float cv[8];
#pragma unroll
  for (int q = 0; q < 8; ++q) cv[q] = acc[q];
  int sel = -1, m = -1;
  if (lane < 8)        { sel = lane;      m = lane;      }
  else if (lane >= 24) { sel = lane - 24; m = lane - 16; }
  float diag = 0.0f;
#pragma unroll
  for (int q = 0; q < 8; ++q) if (sel == q) diag = cv[q];

  if (m >= 0 && (base + m) < count) {
    float v = diag * (SCALE_OBJ * 0.5f);
    atomicAdd(&out[0], v / (float)B_N);            // total_loss == score_loss
    atomicAdd(&out[3], v / (4.0f * (float)B_N));   // score_loss / 4
  }
}

// ------------------------------------------------------------------
// Kernel 4: dense pass over all cells: noobj (m==0), obj (m==1), and the
// sparse xy/wh terms at gt cells (sw>0). Block LDS reduce + atomics.
// ------------------------------------------------------------------
__global__ void k_dense(const float* __restrict__ conf,
                        const float* __restrict__ pxy,
                        const float* __restrict__ pwh,
                        const float* __restrict__ tobj,
                        const float* __restrict__ sw,
                        const float* __restrict__ piou,
                        const int* __restrict__ wrt,
                        float* __restrict__ out) {
  int n = blockIdx.x * blockDim.x + threadIdx.x;
  float p_noobj = 0.f, p_obj = 0.f, p_xy = 0.f, p_wh = 0.f;
  if (n < N_CELL) {
    float m = sw[n];
    float c0 = conf[n];
    if (m == 0.0f) {
      p_noobj = 0.5f * c0 * c0;                         // SCALE_NOOBJ = 1
    } else if (m > 0.0f) {                              // m == 1 cells
      float d = c0 - piou[n];
      p_obj = SCALE_OBJ * 0.5f * d * d;
      // fm index mimics tile(fm_cord,(A,1)): idx = (n % IMG_N) % GRID_N
      int q = n % IMG_N;
      int p = q % GRID_N;
      float fm0 = (float)(p / S_FM);
      float fm1 = (float)(p % S_FM);
      int wr = wrt[n];
      const float* g4 = tobj + wr * 4;
      float tx = g4[0] / DS_R - fm0;
      float ty = g4[1] / DS_R - fm1;
      float tw = g4[2] / DS_R, th = g4[3] / DS_R;
      float px = pxy[2 * n] - fm0, py = pxy[2 * n + 1] - fm1;
      float bx = fmaxf(px, 0.f) - px * tx + log1pf(expf(-fabsf(px)));
      float by = fmaxf(py, 0.f) - py * ty + log1pf(expf(-fabsf(py)));
      p_xy = SCALE_OBJ * 0.5f * m * (bx + by);
      float dw = pwh[2 * n] - tw, dh = pwh[2 * n + 1] - th;
      p_wh = SCALE_OBJ * 0.5f * m * (dw * dw + dh * dh);
    }
    // m == -1: contributes to neither noobj nor obj
  }
  __shared__ float s0[256], s1[256], s2[256], s3[256];
  int tid = threadIdx.x;
  s0[tid] = p_noobj; s1[tid] = p_obj; s2[tid] = p_xy; s3[tid] = p_wh;
  __syncthreads();
  for (int off = 128; off > 0; off >>= 1) {
    if (tid < off) {
      s0[tid] += s0[tid + off]; s1[tid] += s1[tid + off];
      s2[tid] += s2[tid + off]; s3[tid] += s3[tid + off];
    }
    __syncthreads();
  }
  if (tid == 0) {
    float inv = 1.0f / (4.0f * (float)B_N);
    atomicAdd(&out[1], s0[0] * inv);   // noobj_loss / 4
    atomicAdd(&out[2], s1[0] * inv);   // obj_loss / 4
    atomicAdd(&out[4], s2[0] * inv);   // true_loss_xy / 4
    atomicAdd(&out[5], s3[0] * inv);   // true_loss_wh / 4
  }
}

// ------------------------------------------------------------------
extern "C" void kernel_launch(void* const* d_in, const int* in_sizes, int n_in,
                              void* d_out, int out_size, void* d_ws, size_t ws_size,
                              hipStream_t stream) {
  // inputs: 0=epoch 1=conf 2=pred_xy 3=pred_wh 4=cls_score
  //         5=true_object 6=true_label 7=fm_cord 8=anchors
  const float* conf = (const float*)d_in[1];
  const float* pxy  = (const float*)d_in[2];
  const float* pwh  = (const float*)d_in[3];
  const float* cls  = (const float*)d_in[4];
  const float* tobj = (const float*)d_in[5];
  const float* tlab = (const float*)d_in[6];
  const float* anc  = (const float*)d_in[8];
  float* out = (float*)d_out;

  char* ws = (char*)d_ws;
  float* sw      = (float*)(ws);
  float* piou    = (float*)(ws + 4 * (size_t)N_CELL);
  int*   wrt     = (int*)  (ws + 8 * (size_t)N_CELL);
  int*   list    = (int*)  (ws + 12 * (size_t)N_CELL);
  int*   counter = (int*)  (ws + 12 * (size_t)N_CELL + 4096);

  const int BLK = 256;
  const int GRD = N_CELL / BLK;                // 216320 / 256 = 845 exactly

  k_init   <<<GRD, BLK, 0, stream>>>(sw, counter, out);
  k_assign <<<1,   64,  0, stream>>>(pwh, tobj, anc, sw, piou, wrt);
  k_compact<<<GRD, BLK, 0, stream>>>(sw, list, counter);
  k_score  <<<8,   BLK, 0, stream>>>(cls, tlab, list, counter, wrt, out);
  k_dense  <<<GRD, BLK, 0, stream>>>(conf, pxy, pwh, tobj, sw, piou, wrt, out);
}